// Attention_37529424232685
// MI455X (gfx1250) — compile-verified
//
#include <hip/hip_runtime.h>
#include <math.h>

typedef __attribute__((ext_vector_type(2))) float v2f;
typedef __attribute__((ext_vector_type(4))) float v4f;
typedef __attribute__((ext_vector_type(8))) float v8f;

// Problem shape (fixed by the reference):
constexpr int S = 4096;
constexpr int B = 32;
constexpr int H = 1024;
constexpr int WAVES_PER_BLOCK = 4;   // 4 wave32 => 128 threads, 4 s-tiles of 16 per block

// Kernel 1: energies[b, s0+n] = sum_h hidden[b,h] * enc[s0+n, b, h]
// A-matrix (16x4 f32)  = hidden chunk broadcast to all 16 M-rows (LDS broadcast read)
// B-matrix (4x16 f32)  = 16 encoder rows as columns (coalesced-enough b64 stream, NT hint)
// D (16x16 f32)        = every M row identical; lane n VGPR0 = energy[s0+n]
__global__ __launch_bounds__(WAVES_PER_BLOCK * 32)
void Attention_energies_wmma(const float* __restrict__ hid,
                             const float* __restrict__ enc,
                             float* __restrict__ out /* energies [B,S], softmaxed later */)
{
    __shared__ float lh[H];

    const int b = blockIdx.y;
    const int t = threadIdx.x;

    // Stage hidden[b, 0:H] into LDS (128 threads * float4 * 2 passes = 1024 floats)
#pragma unroll
    for (int p = 0; p < H / (WAVES_PER_BLOCK * 32 * 4); ++p) {
        const int idx = t * 4 + p * (WAVES_PER_BLOCK * 32 * 4);
        *(v4f*)(lh + idx) = *(const v4f*)(hid + (size_t)b * H + idx);
    }
    __syncthreads();

    const int wave = t >> 5;
    const int lane = t & 31;
    const int row  = lane & 15;       // N index (s within tile) / M index for A
    const int hi   = (lane >> 4) & 1; // upper-half lanes carry K=2,3 of each 4-chunk
    const int s0   = (blockIdx.x * WAVES_PER_BLOCK + wave) * 16;

    // enc[s, b, h] : row (s0+row, b) is 1024 contiguous floats
    const float* rowp = enc + (size_t)(s0 + row) * (B * H) + (size_t)b * H + 2 * hi;
    const float* hp   = lh + 2 * hi;  // LDS broadcast within each half-wave

    v8f acc0 = {};
    v8f acc1 = {};

#pragma unroll 4
    for (int k0 = 0; k0 < H; k0 += 8) {
        v2f a0 = *(const v2f*)(hp + k0);
        v2f a1 = *(const v2f*)(hp + k0 + 4);
        v2f b0 = __builtin_nontemporal_load((const v2f*)(rowp + k0));
        v2f b1 = __builtin_nontemporal_load((const v2f*)(rowp + k0 + 4));
        // 8 args: (neg_a, A, neg_b, B, c_mod, C, reuse_a, reuse_b)
        acc0 = __builtin_amdgcn_wmma_f32_16x16x4_f32(false, a0, false, b0,
                                                     (short)0, acc0, false, false);
        acc1 = __builtin_amdgcn_wmma_f32_16x16x4_f32(false, a1, false, b1,
                                                     (short)0, acc1, false, false);
    }

    const float e = acc0[0] + acc1[0];   // D[m=0, n=row] on lanes 0..15
    if (hi == 0) {
        out[(size_t)b * S + s0 + row] = e;
    }
}

// Kernel 2: in-place softmax over each of the B rows of length S.
// One 256-thread block per row; 16 elements per thread held in registers.
__global__ __launch_bounds__(256)
void Attention_softmax_rows(float* __restrict__ e)
{
    constexpr int T = 256;
    constexpr int PER = S / T;  // 16

    __shared__ float red[T];
    const int b = blockIdx.x;
    const int t = threadIdx.x;
    float* row = e + (size_t)b * S;

    float v[PER];
    float m = -INFINITY;
#pragma unroll
    for (int i = 0; i < PER; ++i) {
        v[i] = row[t + T * i];
        m = fmaxf(m, v[i]);
    }

    red[t] = m;
    __syncthreads();
#pragma unroll
    for (int off = T / 2; off > 0; off >>= 1) {
        if (t < off) red[t] = fmaxf(red[t], red[t + off]);
        __syncthreads();
    }
    m = red[0];
    __syncthreads();

    float sum = 0.f;
#pragma unroll
    for (int i = 0; i < PER; ++i) {
        v[i] = __expf(v[i] - m);
        sum += v[i];
    }

    red[t] = sum;
    __syncthreads();
#pragma unroll
    for (int off = T / 2; off > 0; off >>= 1) {
        if (t < off) red[t] = red[t] + red[t + off];
        __syncthreads();
    }
    const float inv = 1.0f / red[0];

#pragma unroll
    for (int i = 0; i < PER; ++i) {
        row[t + T * i] = v[i] * inv;
    }
}

extern "C" void kernel_launch(void* const* d_in, const int* in_sizes, int n_in,
                              void* d_out, int out_size, void* d_ws, size_t ws_size,
                              hipStream_t stream)
{
    const float* hid = (const float*)d_in[0];  // [1, B, H] f32
    const float* enc = (const float*)d_in[1];  // [S, B, H] f32
    float* out = (float*)d_out;                // [B, 1, S] f32

    dim3 grid1(S / (16 * WAVES_PER_BLOCK), B); // (64, 32)
    Attention_energies_wmma<<<grid1, WAVES_PER_BLOCK * 32, 0, stream>>>(hid, enc, out);
    Attention_softmax_rows<<<B, 256, 0, stream>>>(out);
}